// Regular_83476984365424
// MI455X (gfx1250) — compile-verified
//
#include <hip/hip_runtime.h>
#include <hip/hip_bf16.h>

typedef __attribute__((ext_vector_type(2))) float v2f;
typedef __attribute__((ext_vector_type(8))) float v8f;

// Sign/scale generator for Hs = (H4 (x) H4) / 4, entries +-0.25.
// H4[a][b] = -1 iff a+b==3; kron sign = XOR of the two digit tests.
__device__ __forceinline__ float hs_val(int i, int j) {
    int n1 = (((i >> 2) + (j >> 2)) == 3);
    int n0 = (((i & 3) + (j & 3)) == 3);
    return (n1 ^ n0) ? -0.25f : 0.25f;
}

// Per 256-element block (viewed as 16x16 row-major X):  Y = Hs * (X * Hs)
// Stage 1: T = X * Hs    (A = X from global, B = Hs const)
// Stage 2: Y = Hs * T    (A = Hs const, B = T via half-wave swaps)
__global__ void had256_wmma_kernel(const float* __restrict__ x,
                                   float* __restrict__ y,
                                   long long nBlocks) {
    const int lane = threadIdx.x & 31;
    const int hi   = lane >> 4;     // 0 for lanes 0-15, 1 for lanes 16-31
    const int r    = lane & 15;

    // Loop-invariant Hs operands.
    // B-layout (4x16 chunk q): reg v, lane l -> Hs[4q + v + 2*hi][l&15]
    // A-layout (16x4 chunk q): lane l -> (Hs[l&15][4q+2*hi], Hs[l&15][4q+2*hi+1])
    v2f HsB[4], HsA[4];
#pragma unroll
    for (int q = 0; q < 4; ++q) {
        HsB[q][0] = hs_val(4 * q + 0 + 2 * hi, r);
        HsB[q][1] = hs_val(4 * q + 1 + 2 * hi, r);
        HsA[q][0] = hs_val(r, 4 * q + 2 * hi + 0);
        HsA[q][1] = hs_val(r, 4 * q + 2 * hi + 1);
    }

    const long long wid     = (long long)blockIdx.x * (blockDim.x >> 5) + (threadIdx.x >> 5);
    const long long wstride = (long long)gridDim.x * (blockDim.x >> 5);

    for (long long blk = wid; blk < nBlocks; blk += wstride) {
        const float* xb = x + blk * 256;

        // Prefetch next iteration's block (speculative; safe past end).
        if (blk + wstride < nBlocks)
            __builtin_prefetch(x + (blk + wstride) * 256 + lane * 8, 0, 1);

        // ---- Stage 1: T = X * Hs  (4 chained K=4 WMMAs) ----
        v8f t = {};
#pragma unroll
        for (int q = 0; q < 4; ++q) {
            // A chunk q in native WMMA A-layout: b64 load of 2 consecutive f32.
            // Streaming data: non-temporal (single-use, keep L2 clean).
            v2f a = __builtin_nontemporal_load(
                        (const v2f*)(xb + r * 16 + 4 * q + 2 * hi));
            t = __builtin_amdgcn_wmma_f32_16x16x4_f32(
                    false, a, false, HsB[q], (short)0, t, false, false);
        }

        // ---- Stage 2: Y = Hs * T ----
        // Build B-layout chunks of T from its C/D layout with half-wave swaps:
        // need lane l<16 -> T[4q+v][l&15], lane l>=16 -> T[4q+v+2][l&15].
        v8f acc = {};
#pragma unroll
        for (int q = 0; q < 4; ++q) {
            v2f b;
#pragma unroll
            for (int v = 0; v < 2; ++v) {
                float lo_v, hi_v;
                if (q < 2) {                         // rows 0..7: live in lanes 0-15
                    lo_v = t[4 * q + v];
                    hi_v = __shfl_xor(t[4 * q + v + 2], 16, 32);
                } else {                             // rows 8..15: live in lanes 16-31
                    lo_v = __shfl_xor(t[4 * q + v - 8], 16, 32);
                    hi_v = t[4 * q + v + 2 - 8];
                }
                b[v] = hi ? hi_v : lo_v;
            }
            acc = __builtin_amdgcn_wmma_f32_16x16x4_f32(
                    false, HsA[q], false, b, (short)0, acc, false, false);
        }

        // ---- Store: acc is Y in C/D layout; reg v holds rows v / v+8.
        // Each store writes two contiguous 64B row segments (coalesced);
        // non-temporal since output is write-once streaming.
        float* yb = y + blk * 256;
#pragma unroll
        for (int v = 0; v < 8; ++v)
            __builtin_nontemporal_store(acc[v], yb + (v + 8 * hi) * 16 + r);
    }
}

extern "C" void kernel_launch(void* const* d_in, const int* in_sizes, int n_in,
                              void* d_out, int out_size, void* d_ws, size_t ws_size,
                              hipStream_t stream) {
    const float* x   = (const float*)d_in[0];
    float*       out = (float*)d_out;

    const long long total   = (long long)in_sizes[0];   // 2*4096*8192
    const long long nBlocks = total / 256;              // 262144 blocks

    const int threads = 256;                            // 8 wave32 per workgroup
    const int wavesPerGroup = threads / 32;
    long long needGroups = (nBlocks + wavesPerGroup - 1) / wavesPerGroup;
    int grid = (int)(needGroups < 2048 ? needGroups : 2048);
    if (grid < 1) grid = 1;

    had256_wmma_kernel<<<grid, threads, 0, stream>>>(x, out, nBlocks);
}